// PointNet2Binary_59717225284093
// MI455X (gfx1250) — compile-verified
//
#include <hip/hip_runtime.h>
#include <hip/hip_bf16.h>

typedef __attribute__((ext_vector_type(2))) float v2f;
typedef __attribute__((ext_vector_type(8))) float v8f;

#define TPB 256

// ---------------------------------------------------------------------------
// Farthest point sampling: one block per batch, sequential over npoint steps.
// ---------------------------------------------------------------------------
__global__ void fps_kernel(const float* __restrict__ xyz, int N, int npoint,
                           float* __restrict__ dist, int* __restrict__ fidx) {
  int b = blockIdx.x;
  xyz += (size_t)b * N * 3;
  dist += (size_t)b * N;
  fidx += (size_t)b * npoint;
  int tid = threadIdx.x, nt = blockDim.x;
  __shared__ float sval[TPB];
  __shared__ int sidx[TPB];
  for (int i = tid; i < N; i += nt) dist[i] = 1e10f;
  __syncthreads();
  int far = 0;
  for (int s = 0; s < npoint; s++) {
    if (tid == 0) fidx[s] = far;
    float cx = xyz[far * 3 + 0], cy = xyz[far * 3 + 1], cz = xyz[far * 3 + 2];
    float best = -1.0f;
    int bi = 0;
    for (int i = tid; i < N; i += nt) {
      float dx = xyz[i * 3 + 0] - cx;
      float dy = xyz[i * 3 + 1] - cy;
      float dz = xyz[i * 3 + 2] - cz;
      float d = dx * dx + dy * dy + dz * dz;
      d = fminf(dist[i], d);
      dist[i] = d;
      if (d > best) { best = d; bi = i; }
    }
    sval[tid] = best; sidx[tid] = bi;
    __syncthreads();
    for (int off = nt >> 1; off > 0; off >>= 1) {
      if (tid < off) {
        if (sval[tid + off] > sval[tid]) {
          sval[tid] = sval[tid + off];
          sidx[tid] = sidx[tid + off];
        }
      }
      __syncthreads();
    }
    far = sidx[0];
    __syncthreads();
  }
}

__global__ void gather_xyz_kernel(const float* __restrict__ xyz, const int* __restrict__ fidx,
                                  float* __restrict__ out, int B, int N, int S) {
  int i = blockIdx.x * blockDim.x + threadIdx.x;
  if (i >= B * S) return;
  int b = i / S;
  int j = fidx[i];
  out[(size_t)i * 3 + 0] = xyz[((size_t)b * N + j) * 3 + 0];
  out[(size_t)i * 3 + 1] = xyz[((size_t)b * N + j) * 3 + 1];
  out[(size_t)i * 3 + 2] = xyz[((size_t)b * N + j) * 3 + 2];
}

// ---------------------------------------------------------------------------
// kNN (k=32): one thread per center, insertion sort of 32 best (ascending d).
// ---------------------------------------------------------------------------
__global__ void knn32_kernel(const float* __restrict__ centers, const float* __restrict__ xyz,
                             int B, int S, int N, int* __restrict__ outidx) {
  int i = blockIdx.x * blockDim.x + threadIdx.x;
  if (i >= B * S) return;
  int b = i / S;
  const float* src = xyz + (size_t)b * N * 3;
  float cx = centers[(size_t)i * 3 + 0];
  float cy = centers[(size_t)i * 3 + 1];
  float cz = centers[(size_t)i * 3 + 2];
  float bd[32];
  int bi[32];
  for (int t = 0; t < 32; t++) { bd[t] = 3.4e38f; bi[t] = 0; }
  for (int j = 0; j < N; j++) {
    float dx = src[j * 3 + 0] - cx;
    float dy = src[j * 3 + 1] - cy;
    float dz = src[j * 3 + 2] - cz;
    float d = dx * dx + dy * dy + dz * dz;
    if (d < bd[31]) {
      int p = 31;
      while (p > 0 && bd[p - 1] > d) {
        bd[p] = bd[p - 1]; bi[p] = bi[p - 1]; p--;
      }
      bd[p] = d; bi[p] = j;
    }
  }
  for (int t = 0; t < 32; t++) outidx[(size_t)i * 32 + t] = bi[t];
}

// ---------------------------------------------------------------------------
// Build grouped tensor [B,S,K, Cpad] (zero padded past 3+Cp):
// centered xyz ++ gathered point features.
// ---------------------------------------------------------------------------
__global__ void group_kernel(const float* __restrict__ xyz, const float* __restrict__ newxyz,
                             const float* __restrict__ points, const int* __restrict__ idx,
                             float* __restrict__ out, int B, int N, int S, int K, int Cp,
                             int Cpad) {
  int C = 3 + Cp;
  size_t tot = (size_t)B * S * K * Cpad;
  size_t gid = (size_t)blockIdx.x * blockDim.x + threadIdx.x;
  if (gid >= tot) return;
  int c = (int)(gid % Cpad);
  size_t t = gid / Cpad;
  int kk = (int)(t % K); t /= K;
  int s = (int)(t % S);
  int b = (int)(t / S);
  float v = 0.0f;
  if (c < C) {
    int j = idx[((size_t)b * S + s) * K + kk];
    if (c < 3)
      v = xyz[((size_t)b * N + j) * 3 + c] - newxyz[((size_t)b * S + s) * 3 + c];
    else
      v = points[((size_t)b * N + j) * Cp + (c - 3)];
  }
  out[gid] = v;
}

// ---------------------------------------------------------------------------
// Zero-pad weights [C,K] -> [C,Kp].
// ---------------------------------------------------------------------------
__global__ void packw_kernel(const float* __restrict__ W, float* __restrict__ Wp,
                             int C, int K, int Kp) {
  size_t tot = (size_t)C * Kp;
  size_t gid = (size_t)blockIdx.x * blockDim.x + threadIdx.x;
  if (gid >= tot) return;
  int k = (int)(gid % Kp);
  int c = (int)(gid / Kp);
  Wp[gid] = (k < K) ? W[(size_t)c * K + k] : 0.0f;
}

// ---------------------------------------------------------------------------
// GEMM + bias via V_WMMA_F32_16X16X4_F32.  H[M,C] = X[M,K] * W[C,K]^T + b.
// Requires K % 4 == 0 (caller pads).  One wave computes a 32x32 output tile
// (2x2 of 16x16); block = 4 waves covering 64x64.  The K loop is software
// pipelined by one stage: fragments for step k+4 are loaded before the WMMAs
// consuming step k's fragments, so load latency overlaps matrix math.
// OOB rows/cols use clamped addresses; their outputs are never stored
// (A row r only affects D row r; B col c only affects D col c).
// A/B frag: lane l, comp j -> K index = k + 2*(l>>4) + j; row/col = l&15.
// C/D frag: vgpr r, lane l -> row = r + 8*(l>>4), col = l&15.
// ---------------------------------------------------------------------------
__global__ void gemm_bias_wmma(const float* __restrict__ X, const float* __restrict__ W,
                               const float* __restrict__ bias, float* __restrict__ H,
                               int M, int K, int C) {
  int lane = threadIdx.x & 31;
  int wave = threadIdx.x >> 5;
  int wm = wave & 1, wn = wave >> 1;
  int mbase = blockIdx.x * 64 + wm * 32;
  int nbase = blockIdx.y * 64 + wn * 32;
  int r15 = lane & 15;
  int half = lane >> 4;
  int kh = half * 2;
  int row0 = mbase + r15, row1 = row0 + 16;
  int col0 = nbase + r15, col1 = col0 + 16;
  int row0c = min(row0, M - 1), row1c = min(row1, M - 1);
  int col0c = min(col0, C - 1), col1c = min(col1, C - 1);
  const float* xr0 = X + (size_t)row0c * K + kh;
  const float* xr1 = X + (size_t)row1c * K + kh;
  const float* wr0 = W + (size_t)col0c * K + kh;
  const float* wr1 = W + (size_t)col1c * K + kh;
  v8f acc00 = {}, acc01 = {}, acc10 = {}, acc11 = {};
  // prologue: load step 0
  v2f a0 = *(const v2f*)(xr0);
  v2f a1 = *(const v2f*)(xr1);
  v2f b0 = *(const v2f*)(wr0);
  v2f b1 = *(const v2f*)(wr1);
  int k = 0;
  for (; k + 4 < K; k += 4) {
    // prefetch step k+4 before consuming step k
    v2f na0 = *(const v2f*)(xr0 + k + 4);
    v2f na1 = *(const v2f*)(xr1 + k + 4);
    v2f nb0 = *(const v2f*)(wr0 + k + 4);
    v2f nb1 = *(const v2f*)(wr1 + k + 4);
    acc00 = __builtin_amdgcn_wmma_f32_16x16x4_f32(false, a0, false, b0, (short)0, acc00, false, false);
    acc01 = __builtin_amdgcn_wmma_f32_16x16x4_f32(false, a0, false, b1, (short)0, acc01, false, false);
    acc10 = __builtin_amdgcn_wmma_f32_16x16x4_f32(false, a1, false, b0, (short)0, acc10, false, false);
    acc11 = __builtin_amdgcn_wmma_f32_16x16x4_f32(false, a1, false, b1, (short)0, acc11, false, false);
    a0 = na0; a1 = na1; b0 = nb0; b1 = nb1;
  }
  // epilogue: consume final step
  acc00 = __builtin_amdgcn_wmma_f32_16x16x4_f32(false, a0, false, b0, (short)0, acc00, false, false);
  acc01 = __builtin_amdgcn_wmma_f32_16x16x4_f32(false, a0, false, b1, (short)0, acc01, false, false);
  acc10 = __builtin_amdgcn_wmma_f32_16x16x4_f32(false, a1, false, b0, (short)0, acc10, false, false);
  acc11 = __builtin_amdgcn_wmma_f32_16x16x4_f32(false, a1, false, b1, (short)0, acc11, false, false);

  float bv0 = bias[col0c];
  float bv1 = bias[col1c];
  bool c0ok = col0 < C, c1ok = col1 < C;
  int mtop = mbase + half * 8;
#pragma unroll
  for (int rr = 0; rr < 8; rr++) {
    int m0 = mtop + rr;
    int m1 = m0 + 16;
    if (m0 < M) {
      if (c0ok) H[(size_t)m0 * C + col0] = acc00[rr] + bv0;
      if (c1ok) H[(size_t)m0 * C + col1] = acc01[rr] + bv1;
    }
    if (m1 < M) {
      if (c0ok) H[(size_t)m1 * C + col0] = acc10[rr] + bv0;
      if (c1ok) H[(size_t)m1 * C + col1] = acc11[rr] + bv1;
    }
  }
}

// ---------------------------------------------------------------------------
// BatchNorm (training-mode batch stats over all M rows), fused scale/shift.
// One block per channel.
// ---------------------------------------------------------------------------
__global__ void bnstats_kernel(const float* __restrict__ H, int M, int C,
                               const float* __restrict__ g, const float* __restrict__ bt,
                               float* __restrict__ scale, float* __restrict__ shift) {
  int c = blockIdx.x;
  int tid = threadIdx.x;
  float s = 0.0f, s2 = 0.0f;
  for (int m = tid; m < M; m += blockDim.x) {
    float v = H[(size_t)m * C + c];
    s += v;
    s2 += v * v;
  }
  __shared__ float rs[TPB], rq[TPB];
  rs[tid] = s; rq[tid] = s2;
  __syncthreads();
  for (int off = blockDim.x >> 1; off > 0; off >>= 1) {
    if (tid < off) { rs[tid] += rs[tid + off]; rq[tid] += rq[tid + off]; }
    __syncthreads();
  }
  if (tid == 0) {
    float mu = rs[0] / (float)M;
    float var = fmaxf(rq[0] / (float)M - mu * mu, 0.0f);
    float inv = rsqrtf(var + 1e-5f);
    float sc = g[c] * inv;
    scale[c] = sc;
    shift[c] = bt[c] - mu * sc;
  }
}

__global__ void bnrelu_kernel(float* __restrict__ H, const float* __restrict__ scale,
                              const float* __restrict__ shift, size_t tot, int C) {
  size_t gid = (size_t)blockIdx.x * blockDim.x + threadIdx.x;
  if (gid >= tot) return;
  int c = (int)(gid % C);
  float v = H[gid] * scale[c] + shift[c];
  H[gid] = fmaxf(v, 0.0f);
}

// ---------------------------------------------------------------------------
// Max-pool over the k neighbors: out[p,c] = max_kk H[(p*K+kk)*C + c].
// ---------------------------------------------------------------------------
__global__ void maxpool_kernel(const float* __restrict__ H, float* __restrict__ out,
                               int P, int K, int C) {
  size_t tot = (size_t)P * C;
  size_t gid = (size_t)blockIdx.x * blockDim.x + threadIdx.x;
  if (gid >= tot) return;
  int c = (int)(gid % C);
  int p = (int)(gid / C);
  float m = -3.4e38f;
  for (int kk = 0; kk < K; kk++)
    m = fmaxf(m, H[((size_t)p * K + kk) * C + c]);
  out[gid] = m;
}

// ---------------------------------------------------------------------------
// 3-NN search + normalized inverse-distance weights.
// ---------------------------------------------------------------------------
__global__ void three_nn_kernel(const float* __restrict__ xyz1, const float* __restrict__ xyz2,
                                int B, int N1, int N2,
                                int* __restrict__ outidx, float* __restrict__ outw) {
  int i = blockIdx.x * blockDim.x + threadIdx.x;
  if (i >= B * N1) return;
  int b = i / N1;
  const float* src = xyz2 + (size_t)b * N2 * 3;
  float cx = xyz1[(size_t)i * 3 + 0];
  float cy = xyz1[(size_t)i * 3 + 1];
  float cz = xyz1[(size_t)i * 3 + 2];
  float bd[3] = {3.4e38f, 3.4e38f, 3.4e38f};
  int bi[3] = {0, 0, 0};
  for (int j = 0; j < N2; j++) {
    float dx = src[j * 3 + 0] - cx;
    float dy = src[j * 3 + 1] - cy;
    float dz = src[j * 3 + 2] - cz;
    float d = dx * dx + dy * dy + dz * dz;
    if (d < bd[2]) {
      int p = 2;
      while (p > 0 && bd[p - 1] > d) {
        bd[p] = bd[p - 1]; bi[p] = bi[p - 1]; p--;
      }
      bd[p] = d; bi[p] = j;
    }
  }
  float w0 = 1.0f / (bd[0] + 1e-8f);
  float w1 = 1.0f / (bd[1] + 1e-8f);
  float w2 = 1.0f / (bd[2] + 1e-8f);
  float wsum = w0 + w1 + w2;
  outw[(size_t)i * 3 + 0] = w0 / wsum;
  outw[(size_t)i * 3 + 1] = w1 / wsum;
  outw[(size_t)i * 3 + 2] = w2 / wsum;
  outidx[(size_t)i * 3 + 0] = bi[0];
  outidx[(size_t)i * 3 + 1] = bi[1];
  outidx[(size_t)i * 3 + 2] = bi[2];
}

__global__ void interp_kernel(const float* __restrict__ points2, const int* __restrict__ idx,
                              const float* __restrict__ w, float* __restrict__ out,
                              int B, int N1, int N2, int C) {
  size_t tot = (size_t)B * N1 * C;
  size_t gid = (size_t)blockIdx.x * blockDim.x + threadIdx.x;
  if (gid >= tot) return;
  int c = (int)(gid % C);
  size_t t = gid / C;
  int i = (int)(t % N1);
  int b = (int)(t / N1);
  const float* p = points2 + (size_t)b * N2 * C;
  size_t base = ((size_t)b * N1 + i) * 3;
  float v = w[base + 0] * p[(size_t)idx[base + 0] * C + c] +
            w[base + 1] * p[(size_t)idx[base + 1] * C + c] +
            w[base + 2] * p[(size_t)idx[base + 2] * C + c];
  out[gid] = v;
}

__global__ void concat_kernel(const float* __restrict__ A, const float* __restrict__ Bp,
                              float* __restrict__ out, size_t Mrows, int C1, int C2) {
  int C = C1 + C2;
  size_t tot = Mrows * C;
  size_t gid = (size_t)blockIdx.x * blockDim.x + threadIdx.x;
  if (gid >= tot) return;
  int c = (int)(gid % C);
  size_t m = gid / C;
  out[gid] = (c < C1) ? A[m * C1 + c] : Bp[m * C2 + (c - C1)];
}

__global__ void head2_kernel(const float* __restrict__ h, const float* __restrict__ w,
                             const float* __restrict__ b, float* __restrict__ out,
                             int Mrows, int Cin, int Cout) {
  size_t tot = (size_t)Mrows * Cout;
  size_t gid = (size_t)blockIdx.x * blockDim.x + threadIdx.x;
  if (gid >= tot) return;
  int o = (int)(gid % Cout);
  size_t m = gid / Cout;
  float s = b[o];
  for (int j = 0; j < Cin; j++) s += h[m * Cin + j] * w[(size_t)o * Cin + j];
  out[gid] = s;
}

// ---------------------------------------------------------------------------
// Host orchestration
// ---------------------------------------------------------------------------
struct Lay { const float *w, *b, *g, *bt; int cin, cout; };

static inline unsigned gblk(size_t tot) { return (unsigned)((tot + TPB - 1) / TPB); }

extern "C" void kernel_launch(void* const* d_in, const int* in_sizes, int n_in,
                              void* d_out, int out_size, void* d_ws, size_t ws_size,
                              hipStream_t stream) {
  (void)in_sizes; (void)n_in; (void)out_size; (void)ws_size;

  const float* x = (const float*)d_in[0];
  int pi = 1;
  auto nx = [&]() { return (const float*)d_in[pi++]; };
  auto mkl = [&](int ci, int co) {
    Lay L; L.w = nx(); L.b = nx(); L.g = nx(); L.bt = nx(); L.cin = ci; L.cout = co; return L;
  };
  Lay sa1[3] = { mkl(3, 64),    mkl(64, 64),   mkl(64, 128) };
  Lay sa2[3] = { mkl(131, 128), mkl(128, 128), mkl(128, 256) };
  Lay sa3[3] = { mkl(259, 256), mkl(256, 256), mkl(256, 512) };
  Lay fp3[2] = { mkl(768, 256), mkl(256, 256) };
  Lay fp2[2] = { mkl(384, 256), mkl(256, 128) };
  Lay fp1[2] = { mkl(128, 128), mkl(128, 128) };
  Lay head1 = mkl(128, 128);
  const float* w2 = nx();
  const float* b2 = nx();

  // ---- carve workspace ----
  float* ws = (float*)d_ws;
  size_t off = 0;
  auto alloc = [&](size_t n) { float* q = ws + off; off += n; return q; };
  float* bufA = alloc((size_t)131072 * 136);
  float* bufB = alloc((size_t)131072 * 128);
  float* dist = alloc((size_t)4 * 8192);
  int* fidx1 = (int*)alloc(4 * 1024);
  int* fidx2 = (int*)alloc(4 * 256);
  int* fidx3 = (int*)alloc(4 * 64);
  float* l1x = alloc((size_t)4 * 1024 * 3);
  float* l2x = alloc((size_t)4 * 256 * 3);
  float* l3x = alloc((size_t)4 * 64 * 3);
  int* idx1 = (int*)alloc((size_t)4 * 1024 * 32);
  int* idx2 = (int*)alloc((size_t)4 * 256 * 32);
  int* idx3 = (int*)alloc((size_t)4 * 64 * 32);
  float* l1p = alloc((size_t)4 * 1024 * 128);
  float* l2p = alloc((size_t)4 * 256 * 256);
  float* l3p = alloc((size_t)4 * 64 * 512);
  float* l0p = alloc((size_t)4 * 8192 * 128);
  int* nnidx = (int*)alloc((size_t)4 * 8192 * 3);
  float* nnw = alloc((size_t)4 * 8192 * 3);
  float* scaleB = alloc(512);
  float* shiftB = alloc(512);
  float* wpad = alloc((size_t)256 * 260);

  auto cbr = [&](const float* Xb, float* Hb, int M, const Lay& L, int Kp) {
    const float* Wuse = L.w;
    if (Kp != L.cin) {
      size_t wtot = (size_t)L.cout * Kp;
      packw_kernel<<<gblk(wtot), TPB, 0, stream>>>(L.w, wpad, L.cout, L.cin, Kp);
      Wuse = wpad;
    }
    dim3 g((unsigned)((M + 63) / 64), (unsigned)((L.cout + 63) / 64));
    gemm_bias_wmma<<<g, 128, 0, stream>>>(Xb, Wuse, L.b, Hb, M, Kp, L.cout);
    bnstats_kernel<<<L.cout, TPB, 0, stream>>>(Hb, M, L.cout, L.g, L.bt, scaleB, shiftB);
    size_t tot = (size_t)M * L.cout;
    bnrelu_kernel<<<gblk(tot), TPB, 0, stream>>>(Hb, scaleB, shiftB, tot, L.cout);
  };

  // ======================== SA1: 8192 -> 1024, k=32 ========================
  fps_kernel<<<4, TPB, 0, stream>>>(x, 8192, 1024, dist, fidx1);
  gather_xyz_kernel<<<gblk(4 * 1024), TPB, 0, stream>>>(x, fidx1, l1x, 4, 8192, 1024);
  knn32_kernel<<<gblk(4 * 1024), TPB, 0, stream>>>(l1x, x, 4, 1024, 8192, idx1);
  group_kernel<<<gblk((size_t)4 * 1024 * 32 * 4), TPB, 0, stream>>>(
      x, l1x, nullptr, idx1, bufA, 4, 8192, 1024, 32, 0, 4);
  int M1 = 4 * 1024 * 32;
  cbr(bufA, bufB, M1, sa1[0], 4);
  cbr(bufB, bufA, M1, sa1[1], 64);
  cbr(bufA, bufB, M1, sa1[2], 64);
  maxpool_kernel<<<gblk((size_t)4 * 1024 * 128), TPB, 0, stream>>>(bufB, l1p, 4 * 1024, 32, 128);

  // ======================== SA2: 1024 -> 256, k=32 =========================
  fps_kernel<<<4, TPB, 0, stream>>>(l1x, 1024, 256, dist, fidx2);
  gather_xyz_kernel<<<gblk(4 * 256), TPB, 0, stream>>>(l1x, fidx2, l2x, 4, 1024, 256);
  knn32_kernel<<<gblk(4 * 256), TPB, 0, stream>>>(l2x, l1x, 4, 256, 1024, idx2);
  group_kernel<<<gblk((size_t)4 * 256 * 32 * 132), TPB, 0, stream>>>(
      l1x, l2x, l1p, idx2, bufA, 4, 1024, 256, 32, 128, 132);
  int M2 = 4 * 256 * 32;
  cbr(bufA, bufB, M2, sa2[0], 132);
  cbr(bufB, bufA, M2, sa2[1], 128);
  cbr(bufA, bufB, M2, sa2[2], 128);
  maxpool_kernel<<<gblk((size_t)4 * 256 * 256), TPB, 0, stream>>>(bufB, l2p, 4 * 256, 32, 256);

  // ======================== SA3: 256 -> 64, k=32 ===========================
  fps_kernel<<<4, TPB, 0, stream>>>(l2x, 256, 64, dist, fidx3);
  gather_xyz_kernel<<<gblk(4 * 64), TPB, 0, stream>>>(l2x, fidx3, l3x, 4, 256, 64);
  knn32_kernel<<<gblk(4 * 64), TPB, 0, stream>>>(l3x, l2x, 4, 64, 256, idx3);
  group_kernel<<<gblk((size_t)4 * 64 * 32 * 260), TPB, 0, stream>>>(
      l2x, l3x, l2p, idx3, bufA, 4, 256, 64, 32, 256, 260);
  int M3 = 4 * 64 * 32;
  cbr(bufA, bufB, M3, sa3[0], 260);
  cbr(bufB, bufA, M3, sa3[1], 256);
  cbr(bufA, bufB, M3, sa3[2], 256);
  maxpool_kernel<<<gblk((size_t)4 * 64 * 512), TPB, 0, stream>>>(bufB, l3p, 4 * 64, 32, 512);

  // ======================== FP3: (l2x <- l3x), 768->256->256 ===============
  three_nn_kernel<<<gblk(4 * 256), TPB, 0, stream>>>(l2x, l3x, 4, 256, 64, nnidx, nnw);
  interp_kernel<<<gblk((size_t)4 * 256 * 512), TPB, 0, stream>>>(l3p, nnidx, nnw, bufB, 4, 256, 64, 512);
  concat_kernel<<<gblk((size_t)4 * 256 * 768), TPB, 0, stream>>>(l2p, bufB, bufA, (size_t)4 * 256, 256, 512);
  cbr(bufA, bufB, 4 * 256, fp3[0], 768);
  cbr(bufB, l2p, 4 * 256, fp3[1], 256);   // l2p now holds fp3 output [1024,256]

  // ======================== FP2: (l1x <- l2x), 384->256->128 ===============
  three_nn_kernel<<<gblk(4 * 1024), TPB, 0, stream>>>(l1x, l2x, 4, 1024, 256, nnidx, nnw);
  interp_kernel<<<gblk((size_t)4 * 1024 * 256), TPB, 0, stream>>>(l2p, nnidx, nnw, bufB, 4, 1024, 256, 256);
  concat_kernel<<<gblk((size_t)4 * 1024 * 384), TPB, 0, stream>>>(l1p, bufB, bufA, (size_t)4 * 1024, 128, 256);
  cbr(bufA, bufB, 4 * 1024, fp2[0], 384);
  cbr(bufB, l1p, 4 * 1024, fp2[1], 256);  // l1p now holds fp2 output [4096,128]

  // ======================== FP1: (x <- l1x), 128->128->128 =================
  three_nn_kernel<<<gblk(4 * 8192), TPB, 0, stream>>>(x, l1x, 4, 8192, 1024, nnidx, nnw);
  interp_kernel<<<gblk((size_t)4 * 8192 * 128), TPB, 0, stream>>>(l1p, nnidx, nnw, bufA, 4, 8192, 1024, 128);
  cbr(bufA, bufB, 4 * 8192, fp1[0], 128);
  cbr(bufB, l0p, 4 * 8192, fp1[1], 128);

  // ======================== Head ===========================================
  cbr(l0p, bufA, 4 * 8192, head1, 128);
  head2_kernel<<<gblk((size_t)4 * 8192 * 2), TPB, 0, stream>>>(bufA, w2, b2, (float*)d_out, 4 * 8192, 128, 2);
}